// CondInst_16990890623098
// MI455X (gfx1250) — compile-verified
//
#include <hip/hip_runtime.h>
#include <math.h>

// CondInst mask head, MI455X (gfx1250, wave32).
//
// B=4, K=100, C=8, H=W=128, P=169. ~2 GFLOP fp32, ~28.5 MB HBM (~1.2us BW
// floor at 23.3 TB/s); the MLP FLOPs would take several us on VALU alone, so
// layers 0 and 1 (288 of 304 FLOP/pixel) run on the matrix pipe, which
// co-executes with VALU (ISA Ch5: XDL WMMA tracked like TRANS ops).
//
// One wave = 2 queries (same batch) x 512 pixels (= 4 image rows x 8 tiles
// of 16 pixels).
// Layer 0 (8<-10): bias + rel terms folded into the C accumulator; the
//   ry-dependent half is hoisted per image row (brow), so the per-tile init
//   is 8 FMAs. Feature part = (16x8)@(8x16) as two K=4 fp32 WMMAs at 100%
//   utilization (M = 2 queries x 8 channels, N = 16 pixels, K = 8).
// Layer 1 (8<-8):  block-diagonal A1 = w1(q0) (+) w1(q1) -> four K=4 fp32
//   WMMAs. D->B operand conversion: every needed value is either in the
//   lane's own h0 register or its lane^16 partner's -> 8x ds_swizzle_b32
//   SWAPX16 (LDS crossbar, no LDS memory) + 8x v_cndmask.
// Layer 2 (1<-8):  8 per-lane FMAs on the C/D layout (WMMA would be 1/8
//   utilized), sigmoid, fully coalesced 4B store.
// ReLU: forced single v_max_num_f32 dst, 0, src via inline asm (the fmed3 /
//   fmaxf forms get instcombine'd into canonicalize+max = 2 ops).
// Inner 8-tile loop unrolled x2 so the scheduler interleaves two
//   independent WMMA/swizzle chains.

typedef __attribute__((ext_vector_type(2))) float v2f;
typedef __attribute__((ext_vector_type(8))) float v8f;

#define BQ 4
#define CC 8
#define KQ 100
#define WW 128
#define HW (128 * 128)
#define PP 169
// param-vector section offsets (match reference slicing)
#define S0 80   // end of w0 (8 x 10)
#define S1 144  // end of w1 (8 x 8)
#define S2 152  // end of w2 (1 x 8); then b0[8], b1[8], b2[1]

#define PIX_PER_WAVE 512
#define ROWS_PER_WAVE 4                       // 512 / 128
#define TILES_PER_ROW 8                       // 128 / 16
#define PAIRS (BQ * KQ / 2)                   // 200
#define BLOCKS_PER_PAIR (HW / PIX_PER_WAVE)   // 32
#define WAVES_PER_BLOCK 8
#define GRID_BLOCKS (PAIRS * BLOCKS_PER_PAIR / WAVES_PER_BLOCK)  // 800

// lane^16 cross-half exchange via the LDS crossbar (no LDS memory traffic).
// ds_swizzle group-of-32 encoding: and=0x1f, or=0, xor=0x10 (SWAPX16)
__device__ __forceinline__ float swz16(float x) {
    return __int_as_float(__builtin_amdgcn_ds_swizzle(__float_as_int(x), 0x401f));
}

// single-instruction ReLU; inline asm so instcombine can't re-expand it into
// canonicalize+max. VOP2: src0 = inline const 0, src1 = VGPR.
__device__ __forceinline__ float relu1(float x) {
    float r;
    asm("v_max_num_f32 %0, 0, %1" : "=v"(r) : "v"(x));
    return r;
}

__global__ __launch_bounds__(256) void condinst_mask_head(
    const float* __restrict__ seg,   // (B, C, H, W)
    const float* __restrict__ cw,    // (B, P, H, W)
    const int*   __restrict__ ind,   // (B, K)
    float*       __restrict__ out)   // (B, K, H, W)
{
    const int lane = threadIdx.x & 31;
    const int wave = blockIdx.x * (blockDim.x >> 5) + (threadIdx.x >> 5);
    const int pair = wave / BLOCKS_PER_PAIR;   // 0..199
    const int pblk = wave % BLOCKS_PER_PAIR;   // 0..31
    const int b  = pair / (KQ / 2);
    const int k0 = (pair % (KQ / 2)) * 2;

    // C/D layout: lanes 0-15 hold rows M=0..7 (query 0), lanes 16-31 hold
    // rows M=8..15 (query 1), column N = lane & 15.
    const int half = lane >> 4;   // query selector for C/D-layout work
    const int n    = lane & 15;   // pixel column within 16-wide tile
    // A layout (16x4 f32): both lane halves cover M = lane&15;
    // VGPR v holds K = 4*kc + 2*(lane>>4) + v.
    const int m   = lane & 15;
    const int qm  = m >> 3;       // query selector for A fragment rows
    const int chm = m & 7;        // output channel for A fragment rows

    // ---- gather per-query parameters (param p lives at cw[base + p*HW]) ----
    const int    ind_h  = ind[b * KQ + k0 + half];
    const int    ind_m  = ind[b * KQ + k0 + qm];
    const size_t base_h = (size_t)b * PP * HW + (size_t)ind_h;
    const size_t base_m = (size_t)b * PP * HW + (size_t)ind_m;

    // Layer-0 A fragments: w0 feature columns (x cols 2..9)
    v2f afrag[2];
#pragma unroll
    for (int kc = 0; kc < 2; ++kc) {
        const int col0 = 2 + kc * 4 + half * 2;
        afrag[kc].x = cw[base_m + (size_t)(chm * (CC + 2) + col0)     * HW];
        afrag[kc].y = cw[base_m + (size_t)(chm * (CC + 2) + col0 + 1) * HW];
    }

    // Layer-1 A fragments: block-diagonal 16x16 = w1(q0) (+) w1(q1).
    // Slot (kc,v) carries K-row k = 4kc + 2*half + v; nonzero iff k is in
    // this M-row's query block. Only 4 w1 loads per lane.
    v2f a1frag[4];
#pragma unroll
    for (int kc = 0; kc < 4; ++kc) {
        float wv[2];
#pragma unroll
        for (int v = 0; v < 2; ++v) {
            const int k = kc * 4 + half * 2 + v;
            wv[v] = ((k >> 3) == qm)
                  ? cw[base_m + (size_t)(S0 + chm * CC + (k & 7)) * HW]
                  : 0.0f;
        }
        a1frag[kc].x = wv[0];
        a1frag[kc].y = wv[1];
    }

    // per-lane weights: rel/bias init + biases + layer 2 (query = half)
    float w0r[CC][2], b0r[CC], b1r[CC], w2r[CC];
#pragma unroll
    for (int i = 0; i < CC; ++i) {
        w0r[i][0] = cw[base_h + (size_t)(i * (CC + 2) + 0) * HW];
        w0r[i][1] = cw[base_h + (size_t)(i * (CC + 2) + 1) * HW];
        b0r[i]    = cw[base_h + (size_t)(S2 + i)            * HW];
        b1r[i]    = cw[base_h + (size_t)(S2 + CC + i)       * HW];
        w2r[i]    = cw[base_h + (size_t)(S1 + i)            * HW];
    }
    const float b2r = cw[base_h + (size_t)(S2 + 2 * CC) * HW];

    const float xi = (float)(ind_h & (WW - 1));
    const float yi = (float)(ind_h >> 7);
    const float inv128 = 1.0f / 128.0f;

    const size_t seg_base = (size_t)b * CC * HW;
    const size_t out_base = (size_t)(b * KQ + k0 + half) * HW;
    const bool lo = (half == 0);

    // rx at tile tt of any row: (xi - (16*tt + n))/128 = rx0 - 0.125*tt
    const float rx0 = (xi - (float)n) * inv128;

#pragma unroll
    for (int row = 0; row < ROWS_PER_WAVE; ++row) {
        const int y = pblk * ROWS_PER_WAVE + row;          // image row
        const float ry = (yi - (float)y) * inv128;

        // hoist the ry-dependent half of the layer-0 C init for this row
        float brow[CC];
#pragma unroll
        for (int i = 0; i < CC; ++i)
            brow[i] = fmaf(w0r[i][1], ry, b0r[i]);

        const int rowpix = y * WW + n;

#pragma unroll 2
        for (int tt = 0; tt < TILES_PER_ROW; ++tt) {
            const int pix = rowpix + tt * 16;
            const float rx = fmaf(-0.125f, (float)tt, rx0);

            // ---- layer 0: C init = brow + w0x * rx (8 FMAs) ----
            v8f acc;
#pragma unroll
            for (int i = 0; i < CC; ++i)
                acc[i] = fmaf(w0r[i][0], rx, brow[i]);

            // feature part: two K=4 fp32 WMMAs; B VGPR v = seg row 4kc+2h+v
#pragma unroll
            for (int kc = 0; kc < 2; ++kc) {
                const int c0 = kc * 4 + half * 2;
                v2f bfrag;
                bfrag.x = seg[seg_base + (size_t)(c0)     * HW + pix];
                bfrag.y = seg[seg_base + (size_t)(c0 + 1) * HW + pix];
                acc = __builtin_amdgcn_wmma_f32_16x16x4_f32(
                    false, afrag[kc], false, bfrag, (short)0, acc, false, false);
            }

            // ReLU -> h0 (1 instr/elem)
            float h0[CC];
#pragma unroll
            for (int i = 0; i < CC; ++i) h0[i] = relu1(acc[i]);

            // ---- layer 1 on WMMA: B from own reg or lane^16 partner ----
            float sw[CC];
#pragma unroll
            for (int i = 0; i < CC; ++i) sw[i] = swz16(h0[i]);

            v2f b1f[4];
            b1f[0].x = lo ? h0[0] : sw[2];   // K rows 0..3
            b1f[0].y = lo ? h0[1] : sw[3];
            b1f[1].x = lo ? h0[4] : sw[6];   // K rows 4..7
            b1f[1].y = lo ? h0[5] : sw[7];
            b1f[2].x = lo ? sw[0] : h0[2];   // K rows 8..11
            b1f[2].y = lo ? sw[1] : h0[3];
            b1f[3].x = lo ? sw[4] : h0[6];   // K rows 12..15
            b1f[3].y = lo ? sw[5] : h0[7];

            v8f acc1;
#pragma unroll
            for (int i = 0; i < CC; ++i) acc1[i] = b1r[i];
#pragma unroll
            for (int kc = 0; kc < 4; ++kc)
                acc1 = __builtin_amdgcn_wmma_f32_16x16x4_f32(
                    false, a1frag[kc], false, b1f[kc], (short)0, acc1, false, false);

            // ---- layer 2 (VALU) + sigmoid, fused ReLU ----
            float logit = b2r;
#pragma unroll
            for (int c = 0; c < CC; ++c)
                logit = fmaf(w2r[c], relu1(acc1[c]), logit);

            out[out_base + pix] = 1.0f / (1.0f + __expf(-logit));
        }
    }
}

extern "C" void kernel_launch(void* const* d_in, const int* in_sizes, int n_in,
                              void* d_out, int out_size, void* d_ws, size_t ws_size,
                              hipStream_t stream) {
    (void)in_sizes; (void)n_in; (void)out_size; (void)d_ws; (void)ws_size;
    const float* seg = (const float*)d_in[0];   // seg_feat    (B,C,H,W) f32
    const float* cw  = (const float*)d_in[1];   // conv_weight (B,P,H,W) f32
    const int*   ind = (const int*)d_in[2];     // ind         (B,K)     i32
    float* out = (float*)d_out;                 // (B,K,H,W) f32

    condinst_mask_head<<<GRID_BLOCKS, 256, 0, stream>>>(seg, cw, ind, out);
}